// SparsemaxLossFunction_31782757991057
// MI455X (gfx1250) — compile-verified
//
#include <hip/hip_runtime.h>
#include <hip/hip_bf16.h>

typedef __attribute__((ext_vector_type(2))) float v2f;
typedef __attribute__((ext_vector_type(8))) float v8f;

#define THREADS 1024

// readlane helper (lane index must be literal/uniform at the call site)
__device__ __forceinline__ float rl(float v, int l) {
  return __builtin_bit_cast(float, __builtin_amdgcn_readlane(__builtin_bit_cast(int, v), l));
}

// Inclusive scan of 256 contiguous LDS floats using the matrix unit:
//   M[r][c] = src[r*16+c];  S = M x U  (U = upper-triangular ones, K=16 done as
//   4 chained V_WMMA_F32_16X16X4_F32), then add per-row prefix offsets gathered
//   via v_readlane from S[:,15].  Returns carry-out (tile total + carry_in).
__device__ __forceinline__ float tile_scan_wmma(const float* __restrict__ src,
                                                float* __restrict__ dst,
                                                float carry, int lane, bool squares) {
  const int half = (lane >= 16) ? 1 : 0;
  const int rr   = half ? (lane - 16) : lane;   // A-row for this lane; also B/D column
  v8f acc = {};
#pragma unroll
  for (int g = 0; g < 4; ++g) {
    // A slice: rows striped across lanes; lanes 0-15 hold K = 4g+0/4g+1,
    // lanes 16-31 hold K = 4g+2 / 4g+3 (per ISA 16x4 f32 A layout).
    const int c0 = 4 * g + (half ? 2 : 0);
    float x0 = src[rr * 16 + c0];
    float x1 = src[rr * 16 + c0 + 1];
    if (squares) { x0 *= x0; x1 *= x1; }
    v2f a; a.x = x0; a.y = x1;
    // B slice of the constant upper-triangular ones matrix: U[k][n] = (k <= n)
    v2f b; b.x = (c0     <= rr) ? 1.0f : 0.0f;
           b.y = (c0 + 1 <= rr) ? 1.0f : 0.0f;
    acc = __builtin_amdgcn_wmma_f32_16x16x4_f32(false, a, false, b, (short)0,
                                                acc, false, false);
  }
  // Row totals live in S[r][15]: vgpr r lane 15 (rows 0-7), lane 31 (rows 8-15).
  float t[16];
#pragma unroll
  for (int j = 0; j < 8; ++j) {
    t[j]     = rl(acc[j], 15);
    t[j + 8] = rl(acc[j], 31);
  }
  float offs[16];
  float run = carry;
#pragma unroll
  for (int r = 0; r < 16; ++r) { offs[r] = run; run += t[r]; }
#pragma unroll
  for (int j = 0; j < 8; ++j) {
    const int r = j + (half ? 8 : 0);
    const float off = half ? offs[j + 8] : offs[j];
    dst[r * 16 + rr] = acc[j] + off;  // D layout: vgpr j = rows j / j+8
  }
  return run;
}

// One workgroup per row. Needs (2*padc + 96) floats of dynamic LDS.
__global__ __launch_bounds__(THREADS)
void sparsemax_row_kernel(const float* __restrict__ X, float* __restrict__ colsum,
                          int* __restrict__ g_supp, float* __restrict__ g_tau,
                          int n, int C, int padc) {
  extern __shared__ float smem[];
  float* a0      = smem;            // padc: row values (sorted in place)
  float* a1      = a0 + padc;       // padc: scan buffer
  float* waveTot = a1 + padc;       // 32
  float* waveOff = waveTot + 32;    // 32
  int*   suppCnt = (int*)(waveOff + 32);

  const int tid    = threadIdx.x;
  const int row    = blockIdx.x;
  const int lane   = tid & 31;
  const int wave   = tid >> 5;
  const int nwaves = THREADS / 32;      // 32
  const int perWave = padc / nwaves;    // 1024 for padc=32768
  const int tiles   = perWave / 256;    // 4
  const size_t rowBase = (size_t)row * (size_t)C;

  // 1) load row, pad to power-of-two with -inf (sorts to the tail)
  for (int i = tid; i < padc; i += THREADS)
    a0[i] = (i < C) ? X[rowBase + i] : -__builtin_huge_valf();
  if (tid == 0) *suppCnt = 0;
  __syncthreads();

  // 2) bitonic sort, descending, entirely in LDS (128KB row fits in 320KB WGP LDS)
  for (unsigned k = 2; k <= (unsigned)padc; k <<= 1) {
    for (unsigned j = k >> 1; j > 0; j >>= 1) {
      for (unsigned i = tid; i < (unsigned)padc; i += THREADS) {
        unsigned ixj = i ^ j;
        if (ixj > i) {
          float va = a0[i], vb = a0[ixj];
          bool up = ((i & k) == 0);
          if (up ? (va < vb) : (va > vb)) { a0[i] = vb; a0[ixj] = va; }
        }
      }
      __syncthreads();
    }
  }
  // zero the -inf pad tail so scans are unaffected
  for (int i = tid; i < padc; i += THREADS)
    if (i >= C) a0[i] = 0.0f;
  __syncthreads();

  const int base = wave * perWave;

  // 3) cumsum of squares via WMMA tile scans -> accumulate column sums in L2
  {
    float carry = 0.0f;
    for (int t = 0; t < tiles; ++t)
      carry = tile_scan_wmma(a0 + base + t * 256, a1 + base + t * 256, carry, lane, true);
    if (lane == 0) waveTot[wave] = carry;
    __syncthreads();
    if (tid == 0) {
      float run = 0.0f;
      for (int w = 0; w < nwaves; ++w) { waveOff[w] = run; run += waveTot[w]; }
    }
    __syncthreads();
    const float off = waveOff[wave];
    for (int kk = lane; kk < perWave; kk += 32) {
      int idx = base + kk;
      if (idx < C) atomicAdd(&colsum[idx], a1[idx] + off);
    }
    __syncthreads();
  }

  // 4) cumsum of values via WMMA tile scans -> support size + tau
  {
    float carry = 0.0f;
    for (int t = 0; t < tiles; ++t)
      carry = tile_scan_wmma(a0 + base + t * 256, a1 + base + t * 256, carry, lane, false);
    if (lane == 0) waveTot[wave] = carry;
    __syncthreads();
    if (tid == 0) {
      float run = 0.0f;
      for (int w = 0; w < nwaves; ++w) { waveOff[w] = run; run += waveTot[w]; }
    }
    __syncthreads();
    const float off = waveOff[wave];
    int cnt = 0;
    for (int kk = lane; kk < perWave; kk += 32) {
      int idx = base + kk;
      float cum = a1[idx] + off;
      a1[idx] = cum;                                // finalize inclusive cumsum
      if (idx < C && (float)(idx + 1) * a0[idx] > cum - 1.0f) ++cnt;
    }
    atomicAdd(suppCnt, cnt);
    __syncthreads();
    if (tid == 0) {
      int s = *suppCnt;                              // support size >= 1 always
      g_supp[row] = s;
      g_tau[row]  = (a1[s - 1] - 1.0f) / (float)s;
    }
  }
}

__global__ void zero_ws_kernel(float* __restrict__ colsum, int C) {
  int i = blockIdx.x * blockDim.x + threadIdx.x;
  if (i < C) colsum[i] = 0.0f;
}

__global__ void finalize_kernel(const float* __restrict__ X, const int* __restrict__ tgt,
                                const float* __restrict__ colsum,
                                const int* __restrict__ supp, const float* __restrict__ tau,
                                float* __restrict__ out, int n, int C) {
  int i = blockIdx.x * blockDim.x + threadIdx.x;
  if (i >= n) return;
  float tv = X[(size_t)i * (size_t)C + (size_t)tgt[i]];
  int s = supp[i];
  float t = tau[i];
  float term = colsum[s - 1] - (float)n * (t * t * (float)s);
  out[i] = -tv + (1.0f + term) * 0.5f;
}

extern "C" void kernel_launch(void* const* d_in, const int* in_sizes, int n_in,
                              void* d_out, int out_size, void* d_ws, size_t ws_size,
                              hipStream_t stream) {
  const float* X      = (const float*)d_in[0];
  const int*   target = (const int*)d_in[1];
  float*       out    = (float*)d_out;

  const int n = in_sizes[1];          // 4096 rows (target count)
  const int C = in_sizes[0] / n;      // 32000 classes
  int padc = 1; while (padc < C) padc <<= 1;   // 32768

  // workspace layout: colsum[padc] (only C used) | supp[n] | tau[n]
  float* colsum = (float*)d_ws;
  int*   supp   = (int*)((char*)d_ws + (size_t)padc * sizeof(float));
  float* tau    = (float*)((char*)supp + (size_t)n * sizeof(int));

  zero_ws_kernel<<<(C + 255) / 256, 256, 0, stream>>>(colsum, C);

  const size_t smemBytes = (size_t)(2 * padc + 96) * sizeof(float);  // ~256.4 KB < 320 KB
  (void)hipFuncSetAttribute((const void*)sparsemax_row_kernel,
                            hipFuncAttributeMaxDynamicSharedMemorySize,
                            (int)smemBytes);
  sparsemax_row_kernel<<<n, THREADS, smemBytes, stream>>>(X, colsum, supp, tau, n, C, padc);

  finalize_kernel<<<(n + 255) / 256, 256, 0, stream>>>(X, target, colsum, supp, tau, out, n, C);
}